// MaxPool2dCustom_44899588112770
// MI455X (gfx1250) — compile-verified
//
#include <hip/hip_runtime.h>

// Max pool 2x2, stride 2, valid padding over NCHW fp32:
//   in : 16 x 64 x 512 x 512  (268,435,456 floats, 1 GiB)
//   out: 16 x 64 x 256 x 256  ( 67,108,864 floats, 256 MiB)
//
// Pure streaming kernel: every input read once, every output written once.
// Strategy (MI455X / gfx1250, wave32):
//   - Each thread computes 4 contiguous outputs along W.
//   - Loads: 2x float4 (b128) from row y0, 2x float4 from row y0+1 -> 64 B in.
//   - Store: 1x float4 (b128) -> 16 B out.
//   - All memory ops carry non-temporal hints (TH=NT) so the 1.25 GiB stream
//     does not pollute the 192 MB L2 (no reuse exists).
//   - 256-thread blocks = 8 wave32 waves; fully coalesced 128-bit accesses.

typedef float v4f __attribute__((ext_vector_type(4)));

#define W_IN   512
#define H_IN   512
#define W_OUT  256
#define H_OUT  256
#define PLANES (16 * 64)        // N*C
#define IN_PLANE  (W_IN * H_IN)     // 262144
#define OUT_PLANE (W_OUT * H_OUT)   // 65536

__global__ __launch_bounds__(256)
void maxpool2x2_s2_nchw_f32(const float* __restrict__ in,
                            float* __restrict__ out) {
    // Flat thread id over (plane, y_out, w_group) where each w_group = 4 outputs.
    unsigned tid = blockIdx.x * 256u + threadIdx.x;

    unsigned wg    = tid & 63u;        // 64 groups of 4 outputs per output row
    unsigned rest  = tid >> 6;
    unsigned y     = rest & (H_OUT - 1u);   // output row 0..255
    unsigned plane = rest >> 8;             // 0..1023 (N*C)

    const float* pin = in
        + (size_t)plane * IN_PLANE
        + (size_t)(2u * y) * W_IN
        + (size_t)wg * 8u;                  // 8 input cols -> 4 outputs

    const v4f* r0 = (const v4f*)pin;                 // row 2y,   cols [8wg .. 8wg+7]
    const v4f* r1 = (const v4f*)(pin + W_IN);        // row 2y+1

    // Non-temporal b128 loads (streaming, no reuse -> bypass L2 retention).
    v4f a = __builtin_nontemporal_load(r0);          // row0 cols 0..3
    v4f b = __builtin_nontemporal_load(r0 + 1);      // row0 cols 4..7
    v4f c = __builtin_nontemporal_load(r1);          // row1 cols 0..3
    v4f d = __builtin_nontemporal_load(r1 + 1);      // row1 cols 4..7

    v4f r;
    r.x = fmaxf(fmaxf(a.x, a.y), fmaxf(c.x, c.y));
    r.y = fmaxf(fmaxf(a.z, a.w), fmaxf(c.z, c.w));
    r.z = fmaxf(fmaxf(b.x, b.y), fmaxf(d.x, d.y));
    r.w = fmaxf(fmaxf(b.z, b.w), fmaxf(d.z, d.w));

    v4f* po = (v4f*)(out
        + (size_t)plane * OUT_PLANE
        + (size_t)y * W_OUT
        + (size_t)wg * 4u);

    __builtin_nontemporal_store(r, po);
}

extern "C" void kernel_launch(void* const* d_in, const int* in_sizes, int n_in,
                              void* d_out, int out_size, void* d_ws, size_t ws_size,
                              hipStream_t stream) {
    const float* x = (const float*)d_in[0];
    float* y = (float*)d_out;

    // 67,108,864 outputs / 4 per thread / 256 per block = 65,536 blocks.
    const unsigned total_threads = (PLANES * OUT_PLANE) / 4u;
    dim3 block(256);
    dim3 grid(total_threads / 256u);

    hipLaunchKernelGGL(maxpool2x2_s2_nchw_f32, grid, block, 0, stream, x, y);
}